// SOTAColorEnhancementLoss_16312285790275
// MI455X (gfx1250) — compile-verified
//
#include <hip/hip_runtime.h>
#include <math.h>

typedef float v2f __attribute__((ext_vector_type(2)));
typedef float v8f __attribute__((ext_vector_type(8)));

// ---------------- problem geometry ----------------
constexpr int BATCH = 8, CHAN = 3, HH = 256, WW = 256;
constexpr int IMG   = HH * WW;          // 65536
constexpr int NPIX  = BATCH * IMG;      // 524288 pixels (per channel-triplet)
constexpr int NIMG  = BATCH * CHAN;     // 24 (b,c) images per tensor
constexpr int NCT   = 2 * NIMG;         // 48: pred images then target images
constexpr int KF    = 129;              // rfft bins along W
constexpr int LD1   = 288;              // stage-1 output width: [cos 144 | sin 144]
constexpr int NT1   = 18;               // 288/16 N-tiles, stage 1
constexpr int NT2   = 9;                // 144/16 N-tiles, stage 2

// ---------------- workspace layout (float offsets) ----------------
constexpr size_t ACC   = 0;                                 // 16 scalar accumulators
constexpr size_t CURV  = 16;                                // 96 cnt | 96 ps | 96 ts
constexpr size_t HISTO = 304;                               // 48 * 64 soft-hist bins
constexpr size_t ZEND  = 3376;                              // zeroed every call
constexpr size_t OB1   = 4096;                              // 256*288 stage-1 DFT matrix
constexpr size_t OA2R  = OB1  + (size_t)256 * 288;          // 256*512
constexpr size_t OA2I  = OA2R + (size_t)256 * 512;          // 256*512
constexpr size_t OY    = OA2I + (size_t)256 * 512;          // 48 * 256*288 stage-1 out
constexpr size_t OMRAW = OY   + (size_t)NCT * 256 * 288;    // raw window mask
constexpr size_t OMTMP = OMRAW + (size_t)NPIX;              // box filter temp
constexpr size_t OMFLT = OMTMP + (size_t)NPIX;              // filtered mask

// ---------------- helpers ----------------
__device__ __forceinline__ float blockReduceSum(float v, float* sdata) {
  int tid = threadIdx.x;
  sdata[tid] = v;
  __syncthreads();
  for (int s = 128; s > 0; s >>= 1) {
    if (tid < s) sdata[tid] += sdata[tid + s];
    __syncthreads();
  }
  float r = sdata[0];
  __syncthreads();
  return r;
}

__device__ __forceinline__ void rgb2lab(float r, float g, float b,
                                        float* L, float* A, float* B) {
  auto lin = [](float x) {
    return x > 0.04045f ? __powf((x + 0.055f) * (1.f / 1.055f), 2.4f)
                        : x * (1.f / 12.92f);
  };
  float rl = lin(r), gl = lin(g), bl = lin(b);
  float X = (0.4124564f * rl + 0.3575761f * gl + 0.1804375f * bl) * (1.f / 0.95047f);
  float Y = (0.2126729f * rl + 0.7151522f * gl + 0.0721750f * bl);
  float Z = (0.0193339f * rl + 0.1191920f * gl + 0.9503041f * bl) * (1.f / 1.08883f);
  auto fc = [](float t) {
    return t > 0.008856452f ? __powf(t, 1.f / 3.f)
                            : t * 7.787037f + 0.13793103f;
  };
  float fx = fc(X), fy = fc(Y), fz = fc(Z);
  *L = 116.f * fy - 16.f;
  *A = 500.f * (fx - fy);
  *B = 200.f * (fy - fz);
}

// ---------------- kernels ----------------
__global__ void k_zero(float* ws) {
  int i = blockIdx.x * 256 + threadIdx.x;
  if (i < (int)ZEND) ws[i] = 0.f;
}

// Build DFT constant matrices on device (deterministic, rebuilt every call).
// B1  [256 x 288]: cols [0,144) = cos(2*pi*r*k/256), cols [144,288) = -sin, zero for k>128
// A2R [256 x 512]: [cos(th) | sin(th)]    (Zr = Cr*Yr - Ci*Yi, Ci = -sin)
// A2I [256 x 512]: [-sin(th) | cos(th)]   (Zi = Ci*Yr + Cr*Yi)
__global__ void k_const(float* ws) {
  int i = blockIdx.x * 256 + threadIdx.x;
  const float STEP = 6.283185307179586f / 256.0f;
  if (i < 256 * 288) {
    int r = i / 288, n = i % 288;
    int k = (n < 144) ? n : n - 144;
    float v = 0.f;
    if (k < KF) {
      float s, c;
      __sincosf((float)((r * k) & 255) * STEP, &s, &c);
      v = (n < 144) ? c : -s;
    }
    ws[OB1 + i] = v;
  } else if (i < 256 * 288 + 256 * 512) {
    int j = i - 256 * 288;
    int m = j / 512, q = j % 512, r = q & 255;
    float s, c;
    __sincosf((float)((m * r) & 255) * STEP, &s, &c);
    ws[OA2R + j] = (q < 256) ? c : s;
  } else if (i < 256 * 288 + 2 * 256 * 512) {
    int j = i - 256 * 288 - 256 * 512;
    int m = j / 512, q = j % 512, r = q & 255;
    float s, c;
    __sincosf((float)((m * r) & 255) * STEP, &s, &c);
    ws[OA2I + j] = (q < 256) ? -s : c;
  }
}

// Stage-1 DFT GEMM: Y[ct] = X[ct] (256x256) * B1 (256x288), one wave per 16x16 tile.
__global__ void k_gemm1(const float* __restrict__ pred,
                        const float* __restrict__ targ,
                        float* __restrict__ ws) {
  int lane = threadIdx.x & 31;
  int gw   = blockIdx.x * 8 + (threadIdx.x >> 5);   // 48*16*18 = 13824 waves exactly
  int ct  = gw / (16 * NT1);
  int rem = gw % (16 * NT1);
  int mt = rem / NT1, nt = rem % NT1;
  const float* X = (ct < NIMG) ? (pred + (size_t)ct * IMG)
                               : (targ + (size_t)(ct - NIMG) * IMG);
  const float* B1 = ws + OB1;
  float* Y = ws + OY + (size_t)ct * (256 * LD1);
  int lr = lane & 15, hi = lane >> 4;
  int arow = mt * 16 + lr;
  int bcol = nt * 16 + lr;
  v8f acc = {0.f, 0.f, 0.f, 0.f, 0.f, 0.f, 0.f, 0.f};
  for (int k0 = 0; k0 < 256; k0 += 4) {
    int ka = k0 + 2 * hi;
    v2f a = {X[arow * 256 + ka], X[arow * 256 + ka + 1]};
    v2f b = {B1[ka * LD1 + bcol], B1[(ka + 1) * LD1 + bcol]};
    acc = __builtin_amdgcn_wmma_f32_16x16x4_f32(false, a, false, b,
                                                (short)0, acc, false, false);
  }
#pragma unroll
  for (int i = 0; i < 8; ++i)
    Y[(size_t)(mt * 16 + i + 8 * hi) * LD1 + bcol] = acc[i];
}

// Stage-2 complex DFT GEMM fused with magnitude + focal-frequency reduction.
// One wave computes the (m,n) 16x16 tile of Zr/Zi for BOTH pred and target,
// then reduces d = | |Zt|-|Zp| | / 256 into sum_d / sum_d2.
__global__ void k_gemm2(float* __restrict__ ws) {
  __shared__ float sred[256];
  int lane = threadIdx.x & 31;
  int gw   = blockIdx.x * 8 + (threadIdx.x >> 5);   // 24*16*9 = 3456 waves exactly
  int c   = gw / (16 * NT2);
  int rem = gw % (16 * NT2);
  int mt = rem / NT2, nt = rem % NT2;
  const float* A2R = ws + OA2R;
  const float* A2I = ws + OA2I;
  const float* Yp = ws + OY + (size_t)c * (256 * LD1);
  const float* Yt = ws + OY + (size_t)(c + NIMG) * (256 * LD1);
  int lr = lane & 15, hi = lane >> 4;
  int arow = mt * 16 + lr;
  int col  = nt * 16 + lr;
  auto Bget = [col](const float* Y, int r) {
    return (r < 256) ? Y[r * LD1 + col] : Y[(r - 256) * LD1 + 144 + col];
  };
  v8f zrp = {0.f,0.f,0.f,0.f,0.f,0.f,0.f,0.f};
  v8f zip = zrp, zrt = zrp, zit = zrp;
  for (int k0 = 0; k0 < 512; k0 += 4) {
    int ka = k0 + 2 * hi;
    v2f ar = {A2R[arow * 512 + ka], A2R[arow * 512 + ka + 1]};
    v2f ai = {A2I[arow * 512 + ka], A2I[arow * 512 + ka + 1]};
    v2f bp = {Bget(Yp, ka), Bget(Yp, ka + 1)};
    v2f bt = {Bget(Yt, ka), Bget(Yt, ka + 1)};
    zrp = __builtin_amdgcn_wmma_f32_16x16x4_f32(false, ar, false, bp, (short)0, zrp, false, false);
    zip = __builtin_amdgcn_wmma_f32_16x16x4_f32(false, ai, false, bp, (short)0, zip, false, false);
    zrt = __builtin_amdgcn_wmma_f32_16x16x4_f32(false, ar, false, bt, (short)0, zrt, false, false);
    zit = __builtin_amdgcn_wmma_f32_16x16x4_f32(false, ai, false, bt, (short)0, zit, false, false);
  }
  float vmask = (col < KF) ? 1.f : 0.f;   // only k=0..128 exist in rfft2 output
  float sd = 0.f, sd2 = 0.f;
#pragma unroll
  for (int i = 0; i < 8; ++i) {
    float mp = sqrtf(zrp[i] * zrp[i] + zip[i] * zip[i]) * (1.f / 256.f);
    float mt_ = sqrtf(zrt[i] * zrt[i] + zit[i] * zit[i]) * (1.f / 256.f);
    float d = fabsf(mt_ - mp) * vmask;
    sd += d;
    sd2 += d * d;
  }
  float t1 = blockReduceSum(sd, sred);
  if (threadIdx.x == 0) atomicAdd(&ws[1], t1);
  float t2 = blockReduceSum(sd2, sred);
  if (threadIdx.x == 0) atomicAdd(&ws[2], t2);
}

// Fused pointwise pass: L1, LAB, curve segment sums, window raw mask.
__global__ void k_pix(const float* __restrict__ pred,
                      const float* __restrict__ targ,
                      const float* __restrict__ inp,
                      float* __restrict__ ws) {
  __shared__ float s_curve[288];
  __shared__ float sred[256];
  int tid = threadIdx.x;
  for (int j = tid; j < 288; j += 256) s_curve[j] = 0.f;
  __syncthreads();
  int idx = blockIdx.x * 256 + tid;            // < 524288
  int b = idx >> 16;
  int hw = idx & 65535;
  size_t base = (size_t)b * 3 * IMG + hw;
  float p[3], t[3], x[3];
#pragma unroll
  for (int c = 0; c < 3; ++c) {
    p[c] = pred[base + (size_t)c * IMG];
    t[c] = targ[base + (size_t)c * IMG];
    x[c] = inp[base + (size_t)c * IMG];
  }
  float l1v = fabsf(p[0] - t[0]) + fabsf(p[1] - t[1]) + fabsf(p[2] - t[2]);
  float Lp, Ap, Bp, Lt, At, Bt;
  rgb2lab(p[0], p[1], p[2], &Lp, &Ap, &Bp);
  rgb2lab(t[0], t[1], t[2], &Lt, &At, &Bt);
  float dL = fabsf(Lp - Lt);
  float dab = fabsf(Ap - At) + fabsf(Bp - Bt);
  // target HSV window mask
  float mx = fmaxf(t[0], fmaxf(t[1], t[2]));
  float mn = fminf(t[0], fminf(t[1], t[2]));
  float sat = (mx - mn + 1e-7f) / (mx + 1e-7f);
  ws[OMRAW + idx] = (mx > 0.6f && sat > 0.2f) ? 1.f : 0.f;
  // curve bins
#pragma unroll
  for (int c = 0; c < 3; ++c) {
    int bin = (int)(x[c] * 32.f);
    bin = bin < 0 ? 0 : (bin > 31 ? 31 : bin);
    int seg = c * 32 + bin;
    atomicAdd(&s_curve[seg], 1.f);
    atomicAdd(&s_curve[96 + seg], p[c]);
    atomicAdd(&s_curve[192 + seg], t[c]);
  }
  float r1 = blockReduceSum(l1v, sred);
  if (tid == 0) atomicAdd(&ws[0], r1);
  float r2 = blockReduceSum(dL, sred);
  if (tid == 0) atomicAdd(&ws[3], r2);
  float r3 = blockReduceSum(dab, sred);
  if (tid == 0) atomicAdd(&ws[4], r3);
  for (int j = tid; j < 288; j += 256) atomicAdd(&ws[CURV + j], s_curve[j]);
}

// Multiscale: one thread per 4x4 full-res block -> 4 half diffs + 1 quarter diff.
__global__ void k_ms(const float* __restrict__ pred,
                     const float* __restrict__ targ,
                     float* __restrict__ ws) {
  __shared__ float sred[256];
  int idx = blockIdx.x * 256 + threadIdx.x;    // < 98304
  int img = idx >> 12;                          // (b*3+c)
  int r2 = idx & 4095;
  int qh = r2 >> 6, qw = r2 & 63;
  const float* P = pred + (size_t)img * IMG;
  const float* T = targ + (size_t)img * IMG;
  float sh = 0.f, pq = 0.f, tq = 0.f;
#pragma unroll
  for (int a = 0; a < 2; ++a)
#pragma unroll
    for (int b = 0; b < 2; ++b) {
      int h0 = (qh * 2 + a) * 2, w0 = (qw * 2 + b) * 2;
      float ph = 0.25f * (P[h0 * 256 + w0] + P[h0 * 256 + w0 + 1] +
                          P[(h0 + 1) * 256 + w0] + P[(h0 + 1) * 256 + w0 + 1]);
      float th = 0.25f * (T[h0 * 256 + w0] + T[h0 * 256 + w0 + 1] +
                          T[(h0 + 1) * 256 + w0] + T[(h0 + 1) * 256 + w0 + 1]);
      sh += fabsf(ph - th);
      pq += ph;
      tq += th;
    }
  float sq = fabsf(0.25f * (pq - tq));
  float t1 = blockReduceSum(sh, sred);
  if (threadIdx.x == 0) atomicAdd(&ws[5], t1);
  float t2 = blockReduceSum(sq, sred);
  if (threadIdx.x == 0) atomicAdd(&ws[6], t2);
}

// Soft histogram: register-private 64-bin accumulators, flush via LDS.
__global__ void k_hist(const float* __restrict__ pred,
                       const float* __restrict__ targ,
                       float* __restrict__ ws) {
  int combo = blockIdx.x >> 3;                 // 0..47
  int split = blockIdx.x & 7;
  const float* img = (combo < NIMG) ? (pred + (size_t)combo * IMG)
                                    : (targ + (size_t)(combo - NIMG) * IMG);
  float h[64];
#pragma unroll
  for (int j = 0; j < 64; ++j) h[j] = 0.f;
  int end = (split + 1) * 8192;
  for (int p = split * 8192 + threadIdx.x; p < end; p += 256) {
    float x = img[p];
#pragma unroll
    for (int j = 0; j < 64; ++j) {
      float d = x - (float)j * (1.f / 63.f);
      h[j] += __expf(-2048.f * d * d);         // 1/(2*bw^2), bw = 1/64
    }
  }
  __shared__ float sh[64];
  if (threadIdx.x < 64) sh[threadIdx.x] = 0.f;
  __syncthreads();
#pragma unroll
  for (int j = 0; j < 64; ++j) {
    int jj = (j + threadIdx.x) & 63;           // rotate to spread LDS banks
    atomicAdd(&sh[jj], h[jj]);
  }
  __syncthreads();
  if (threadIdx.x < 64)
    atomicAdd(&ws[HISTO + (size_t)combo * 64 + threadIdx.x], sh[threadIdx.x]);
}

// Normalize + cumsum + EMD per (b,c) image pair.
__global__ void k_histfin(float* __restrict__ ws) {
  if (threadIdx.x != 0) return;
  int c = blockIdx.x;
  const float* hp = ws + HISTO + (size_t)c * 64;
  const float* ht = ws + HISTO + (size_t)(c + NIMG) * 64;
  float sp = 0.f, st = 0.f;
  for (int j = 0; j < 64; ++j) { sp += hp[j]; st += ht[j]; }
  float ip = 1.f / (sp + 1e-7f), it = 1.f / (st + 1e-7f);
  float cp = 0.f, ct = 0.f, hl = 0.f;
  for (int j = 0; j < 64; ++j) {
    cp += hp[j] * ip;
    ct += ht[j] * it;
    hl += fabsf(cp - ct);
  }
  atomicAdd(&ws[7], hl);
}

// Separable 11x11 box filter (count_include_pad) on the window mask.
__global__ void k_boxh(float* __restrict__ ws) {
  int idx = blockIdx.x * 256 + threadIdx.x;    // < 524288
  int w = idx & 255;
  int rowbase = idx - w;
  float s = 0.f;
  for (int d = -5; d <= 5; ++d) {
    int wn = w + d;
    if (wn >= 0 && wn < 256) s += ws[OMRAW + rowbase + wn];
  }
  ws[OMTMP + idx] = s;
}

__global__ void k_boxv(float* __restrict__ ws) {
  int idx = blockIdx.x * 256 + threadIdx.x;
  int hw = idx & 65535;
  int h = hw >> 8, w = hw & 255;
  int imgbase = idx - hw;
  float s = 0.f;
  for (int d = -5; d <= 5; ++d) {
    int hn = h + d;
    if (hn >= 0 && hn < 256) s += ws[OMTMP + imgbase + hn * 256 + w];
  }
  ws[OMFLT + idx] = s * (1.f / 121.f);
}

// Window-boost sums: mask total, masked |pred-target|, masked saturation diff.
__global__ void k_win(const float* __restrict__ pred,
                      const float* __restrict__ targ,
                      float* __restrict__ ws) {
  __shared__ float sred[256];
  int idx = blockIdx.x * 256 + threadIdx.x;
  int b = idx >> 16;
  int hw = idx & 65535;
  float m = ws[OMFLT + idx];
  size_t base = (size_t)b * 3 * IMG + hw;
  float p0 = pred[base], p1 = pred[base + IMG], p2 = pred[base + 2 * IMG];
  float t0 = targ[base], t1 = targ[base + IMG], t2 = targ[base + 2 * IMG];
  float sa = fabsf(p0 - t0) + fabsf(p1 - t1) + fabsf(p2 - t2);
  float pmx = fmaxf(p0, fmaxf(p1, p2)), pmn = fminf(p0, fminf(p1, p2));
  float tmx = fmaxf(t0, fmaxf(t1, t2)), tmn = fminf(t0, fminf(t1, t2));
  float ps = (pmx - pmn + 1e-7f) / (pmx + 1e-7f);
  float ts = (tmx - tmn + 1e-7f) / (tmx + 1e-7f);
  float v1 = blockReduceSum(m, sred);
  if (threadIdx.x == 0) atomicAdd(&ws[8], v1);
  float v2 = blockReduceSum(sa * m, sred);
  if (threadIdx.x == 0) atomicAdd(&ws[9], v2);
  float v3 = blockReduceSum(fabsf(ps - ts) * m, sred);
  if (threadIdx.x == 0) atomicAdd(&ws[10], v3);
}

__global__ void k_final(const float* __restrict__ ws, float* __restrict__ out) {
  if (threadIdx.x != 0 || blockIdx.x != 0) return;
  float l1 = ws[0] / (float)(NPIX * 3);
  float nf = (float)NIMG * 256.f * (float)KF;
  float md = ws[1] / nf;
  float ff = (ws[2] / nf) / (md + 1e-8f);
  float l_loss = ws[3] / (float)NPIX / 100.f;
  float ab_loss = ws[4] / (float)(2 * NPIX) / 128.f;
  float lab = l_loss + 2.f * ab_loss;
  float ms = (l1 + ws[5] / (float)(NIMG * 128 * 128) +
              ws[6] / (float)(NIMG * 64 * 64)) / 3.f;
  float curve = 0.f;
  for (int j = 0; j < 96; ++j) {
    float cnt = ws[CURV + j];
    float pc = cnt > 0.f ? ws[CURV + 96 + j] / fmaxf(cnt, 1.f) : 0.f;
    float tc = cnt > 0.f ? ws[CURV + 192 + j] / fmaxf(cnt, 1.f) : 0.f;
    curve += fabsf(pc - tc);
  }
  curve *= (1.f / 96.f);
  float hl = ws[7] / (float)(NIMG * 64);
  float msum = ws[8];
  float wloss = ws[9] / (msum * 3.f + 1e-7f);
  float satl = ws[10] / (msum + 1e-7f);
  float wb = wloss + 0.5f * satl;
  out[0] = 1.0f * l1 + 0.3f * ff + 0.4f * lab + 0.3f * ms +
           0.4f * curve + 0.2f * hl + 0.5f * wb;
}

// ---------------- launcher ----------------
extern "C" void kernel_launch(void* const* d_in, const int* in_sizes, int n_in,
                              void* d_out, int out_size, void* d_ws, size_t ws_size,
                              hipStream_t stream) {
  (void)in_sizes; (void)n_in; (void)out_size; (void)ws_size;
  const float* pred = (const float*)d_in[0];
  const float* targ = (const float*)d_in[1];
  const float* inp  = (const float*)d_in[2];
  float* ws = (float*)d_ws;
  float* out = (float*)d_out;

  k_zero<<<(int)((ZEND + 255) / 256), 256, 0, stream>>>(ws);
  k_const<<<(256 * 288 + 2 * 256 * 512 + 255) / 256, 256, 0, stream>>>(ws);
  k_gemm1<<<(NCT * 16 * NT1) / 8, 256, 0, stream>>>(pred, targ, ws);   // 1728 blocks
  k_gemm2<<<(NIMG * 16 * NT2) / 8, 256, 0, stream>>>(ws);              // 432 blocks
  k_pix<<<NPIX / 256, 256, 0, stream>>>(pred, targ, inp, ws);
  k_ms<<<(NIMG * 64 * 64) / 256, 256, 0, stream>>>(pred, targ, ws);
  k_hist<<<NCT * 8, 256, 0, stream>>>(pred, targ, ws);
  k_histfin<<<NIMG, 32, 0, stream>>>(ws);
  k_boxh<<<NPIX / 256, 256, 0, stream>>>(ws);
  k_boxv<<<NPIX / 256, 256, 0, stream>>>(ws);
  k_win<<<NPIX / 256, 256, 0, stream>>>(pred, targ, ws);
  k_final<<<1, 32, 0, stream>>>(ws, out);
}